// MaskRCNN_85538568667206
// MI455X (gfx1250) — compile-verified
//
#include <hip/hip_runtime.h>
#include <hip/hip_bf16.h>
#include <stdint.h>

#define A_MAX 20
#define IMG_C 1024

typedef float vf2 __attribute__((ext_vector_type(2)));
typedef float vf4 __attribute__((ext_vector_type(4)));

// ---- compile-time level geometry ------------------------------------------
constexpr int hw_of(int lv) { int h = IMG_C >> lv; return h * h; }
constexpr int n0_of(int lv) { int s = 0; for (int i = 2; i < lv; ++i) s += 3 * hw_of(i); return s; }
constexpr int N_of()        { int s = 0; for (int i = 2; i < 7;  ++i) s += 3 * hw_of(i); return s; }

// ---- CDNA5 async global->LDS broadcast (ASYNCcnt path) --------------------
// global_load_async_to_lds_b128 vdst(LDS byte addr), vaddr(64-bit addr), off
__device__ __forceinline__ void async_b128_to_lds(const float* gsrc, float* ldst) {
  // Low 32 bits of a generic pointer to LDS == LDS byte offset (ISA aperture rule).
  uint32_t lds_addr = (uint32_t)(uintptr_t)ldst;
  asm volatile("global_load_async_to_lds_b128 %0, %1, off"
               :: "v"(lds_addr), "v"(gsrc)
               : "memory");
}

__device__ __forceinline__ void wait_async0() {
#if __has_builtin(__builtin_amdgcn_s_wait_asynccnt)
  __builtin_amdgcn_s_wait_asynccnt(0);
#else
  asm volatile("s_wait_asynccnt 0" ::: "memory");
#endif
}

// ---- per-level RPN proposal + label kernel --------------------------------
// Thread -> (anchor type a, y, x); blockIdx.y = batch b. LEVEL is compile-time
// so all index math lowers to shifts/masks and scale constants fold.
template <int LEVEL>
__global__ __launch_bounds__(256) void rpn_level_kernel(
    const float* __restrict__ cs, const float* __restrict__ bp,
    const float* __restrict__ gt, const int* __restrict__ gtc,
    const int* __restrict__ imh_p, const int* __restrict__ imw_p,
    float* __restrict__ out_scores, float* __restrict__ out_boxes,
    float* __restrict__ out_anchors, float* __restrict__ out_labels)
{
  constexpr int H    = IMG_C >> LEVEL;        // h == w
  constexpr int hw   = H * H;
  constexpr int tot  = 3 * hw;
  constexpr int n0   = n0_of(LEVEL);
  constexpr int N    = N_of();
  constexpr int numA = A_MAX;
  constexpr float s    = (float)(1 << LEVEL);
  constexpr float half = 0.5f * (s - 1.0f);
  constexpr float inv  = 1.0f / (float)(1 << (6 - LEVEL));

  __shared__ alignas(16) float gts[A_MAX * 8];  // y1,x1,y2,x2,area, pad (32B stride)

  const unsigned b = blockIdx.y;
  const unsigned t = threadIdx.x;

  // Broadcast this batch's GT boxes into LDS with the async DMA path.
  if (t < numA) {
    async_b128_to_lds(gt + (b * numA + t) * 4, &gts[t * 8]);
  }
  wait_async0();
  __syncthreads();
  if (t < numA) {  // precompute GT areas once per block
    const float gy1 = gts[t * 8 + 0], gx1 = gts[t * 8 + 1];
    const float gy2 = gts[t * 8 + 2], gx2 = gts[t * 8 + 3];
    gts[t * 8 + 4] = (gy2 - gy1) * (gx2 - gx1);
  }
  __syncthreads();

  const unsigned idx = blockIdx.x * blockDim.x + t;
  if (idx >= (unsigned)tot) return;

  const unsigned a   = idx / hw;           // power-of-two: shift
  const unsigned rem = idx & (hw - 1);
  const unsigned y   = rem / H;            // power-of-two: shift
  const unsigned x   = rem & (H - 1);

  // Anchor center: v -> 2^i * v + 0.5*(2^i - 1)
  const float cy = fmaf(s, (float)y, half);
  const float cx = fmaf(s, (float)x, half);
  const float ha = (float)imh_p[0] * inv;
  const float wa = (float)imw_p[0] * inv;
  const float hh = (a == 2) ? 2.0f * ha : ha;   // heights [ha, ha, 2ha]
  const float ww = (a == 0) ? 2.0f * wa : wa;   // widths  [2wa, wa, wa]

  const unsigned n = n0 + idx;

  // Scores: two coalesced channel-plane loads (streaming -> nontemporal).
  const size_t base_cs = ((size_t)b * 6 + (size_t)a * 2) * hw + rem;
  const float s0 = __builtin_nontemporal_load(cs + base_cs);
  const float s1 = __builtin_nontemporal_load(cs + base_cs + hw);
  {
    vf2 sc = {s0, s1};
    __builtin_nontemporal_store(sc, (vf2*)out_scores + ((size_t)b * N + n));
  }

  // Box deltas: four coalesced loads.
  const size_t base_bp = ((size_t)b * 12 + (size_t)a * 4) * hw + rem;
  const float dy = __builtin_nontemporal_load(bp + base_bp);
  const float dx = __builtin_nontemporal_load(bp + base_bp + hw);
  const float dh = __builtin_nontemporal_load(bp + base_bp + 2 * (size_t)hw);
  const float dw = __builtin_nontemporal_load(bp + base_bp + 3 * (size_t)hw);

  const float cyc = fmaf(dy, hh, cy);
  const float cxc = fmaf(dx, ww, cx);
  const float h2  = hh * __expf(dh);
  const float w2  = ww * __expf(dw);

  {
    vf4 bx = {fmaf(-0.5f, h2, cyc), fmaf(-0.5f, w2, cxc),
              fmaf( 0.5f, h2, cyc), fmaf( 0.5f, w2, cxc)};
    __builtin_nontemporal_store(bx, (vf4*)out_boxes + ((size_t)b * N + n));
    vf4 an = {cy, cx, hh, ww};
    __builtin_nontemporal_store(an, (vf4*)out_anchors + ((size_t)b * N + n));
  }

  // Labels: any(IoU >= 0.5) over valid GT boxes (LDS broadcast reads).
  const float y1 = fmaf(-0.5f, hh, cy), y2 = y1 + hh;
  const float x1 = fmaf(-0.5f, ww, cx), x2 = x1 + ww;
  const float area = hh * ww;
  int cnt = gtc[b];
  if (cnt > numA) cnt = numA;
  int lab = 0;
  for (int j = 0; j < cnt; ++j) {
    const float gy1 = gts[j * 8 + 0], gx1 = gts[j * 8 + 1];
    const float gy2 = gts[j * 8 + 2], gx2 = gts[j * 8 + 3];
    const float ga  = gts[j * 8 + 4];
    const float yy1 = fminf(fmaxf(y1, gy1), gy2);
    const float yy2 = fminf(fmaxf(y2, gy1), gy2);
    const float xx1 = fminf(fmaxf(x1, gx1), gx2);
    const float xx2 = fminf(fmaxf(x2, gx1), gx2);
    const float inter = (yy2 - yy1) * (xx2 - xx1);
    // iou >= 0.5  <=>  inter >= 0.5*(area+ga-inter)  <=>  3*inter >= area+ga
    lab |= (3.0f * inter >= area + ga) ? 1 : 0;
  }
  __builtin_nontemporal_store((float)lab, out_labels + ((size_t)b * N + n));
}

extern "C" void kernel_launch(void* const* d_in, const int* in_sizes, int n_in,
                              void* d_out, int out_size, void* d_ws, size_t ws_size,
                              hipStream_t stream) {
  (void)in_sizes; (void)n_in; (void)out_size; (void)d_ws; (void)ws_size;
  constexpr int Bc = 2;
  constexpr int N  = N_of();

  float* out         = (float*)d_out;
  float* out_scores  = out;
  float* out_boxes   = out + (size_t)Bc * N * 2;
  float* out_anchors = out + (size_t)Bc * N * 6;
  float* out_labels  = out + (size_t)Bc * N * 10;

  // setup_inputs dict order: cs2,bp2,cs3,bp3,cs4,bp4,cs5,bp5,cs6,bp6,gt,gtc,imh,imw
  const float* gt  = (const float*)d_in[10];
  const int*   gtc = (const int*)d_in[11];
  const int*   imh = (const int*)d_in[12];
  const int*   imw = (const int*)d_in[13];

#define LAUNCH_LEVEL(LV, IN_IDX)                                               \
  do {                                                                         \
    constexpr int tot = 3 * hw_of(LV);                                         \
    dim3 grid((tot + 255) / 256, Bc);                                          \
    hipLaunchKernelGGL((rpn_level_kernel<LV>), grid, dim3(256), 0, stream,     \
                       (const float*)d_in[IN_IDX], (const float*)d_in[IN_IDX + 1], \
                       gt, gtc, imh, imw,                                      \
                       out_scores, out_boxes, out_anchors, out_labels);        \
  } while (0)

  LAUNCH_LEVEL(2, 0);
  LAUNCH_LEVEL(3, 2);
  LAUNCH_LEVEL(4, 4);
  LAUNCH_LEVEL(5, 6);
  LAUNCH_LEVEL(6, 8);
#undef LAUNCH_LEVEL
}